// CrossWaveletTransform_37366215475274
// MI455X (gfx1250) — compile-verified
//
#include <hip/hip_runtime.h>
#include <math.h>

// ---------------------------------------------------------------------------
// CWT (pywt cmor, real part) as a scale-blocked fp32 WMMA matmul on gfx1250.
//
//   out[b,s,l] = sum_m x[b, l+m] * G[s-1, m+1025]
//   G[s-1, cm] = -sqrt(s) * (F(k-1) - F(k)),  k = cm + 8s - 1024, k in [0,16s+1]
//   F(k) = int_psi[ floor(k / (s*step)) ] for k in [0,16s], else 0
//
// Main kernel: workgroup = 32 scales (pair of 16-scale blocks) x 512
// positions. Each wave computes a 2x4 grid of 16x16 tiles (8 accumulators):
// 8 wmmas share 2 A-fragment global loads and 4 B-fragment LDS loads,
// halving memory traffic per matrix op vs 1x4 tiling. The pair runs at the
// larger block's K extent; the smaller block's filter rows are zero there.
// ---------------------------------------------------------------------------

typedef float v2f __attribute__((ext_vector_type(2)));
typedef float v8f __attribute__((ext_vector_type(8)));

#define NSCALES 128
#define GSTRIDE 2052          // aligned filter row length (multiple of 4)
#define SIGLEN  8192
#define NBATCH  32
#define PREC_N  1024
#define WGPOS   512           // positions per workgroup = 8 waves * 4 * 16

// ---------------------------------------------------------------------------
// Kernel 1: integrated wavelet table int_psi[1024] (double precision scan).
// ---------------------------------------------------------------------------
__global__ void cwt_int_psi_kernel(float* __restrict__ int_psi) {
  __shared__ double sp[PREC_N];
  const int i = threadIdx.x;                 // blockDim.x == 1024
  const double d    = 16.0 / 1023.0;         // linspace delta
  const double step = (-8.0 + d) + 8.0;      // x[1]-x[0] as numpy computes it
  const double xi   = (double)i * d - 8.0;   // x[i] = i*delta + start
  const double INV_SQRT_PI = 0.5641895835477563;
  const double TWO_PI      = 6.283185307179586;
  sp[i] = INV_SQRT_PI * exp(-xi * xi) * cos(TWO_PI * xi);
  __syncthreads();
  for (int off = 1; off < PREC_N; off <<= 1) {
    double t = (i >= off) ? sp[i - off] : 0.0;
    __syncthreads();
    sp[i] += t;
    __syncthreads();
  }
  int_psi[i] = (float)(sp[i] * step);
}

// ---------------------------------------------------------------------------
// Kernel 2: build aligned, pre-scaled filter bank G[128][GSTRIDE].
// ---------------------------------------------------------------------------
__device__ __forceinline__ float cwt_fsample(const float* __restrict__ int_psi,
                                             int k, int s, double sstep) {
  if (k < 0 || k > 16 * s) return 0.0f;
  int j = (int)floor((double)k / sstep);     // numpy: floor(i / (s*step))
  if (j > PREC_N - 1) j = PREC_N - 1;        // reference drops j >= 1024
  return int_psi[j];
}

__global__ void cwt_build_filters_kernel(const float* __restrict__ int_psi,
                                         float* __restrict__ G) {
  int idx = blockIdx.x * blockDim.x + threadIdx.x;
  if (idx >= NSCALES * GSTRIDE) return;
  int srow = idx / GSTRIDE;
  int cm   = idx - srow * GSTRIDE;           // aligned column, m = cm - 1025
  int s    = srow + 1;
  int k    = cm + 8 * s - 1024;
  float val = 0.0f;
  if (k >= 0 && k <= 16 * s + 1) {
    const double d    = 16.0 / 1023.0;
    const double step = (-8.0 + d) + 8.0;
    const double sstep = (double)s * step;
    float fkm1 = cwt_fsample(int_psi, k - 1, s, sstep);
    float fk   = cwt_fsample(int_psi, k,     s, sstep);
    val = -sqrtf((float)s) * (fkm1 - fk);    // fold -sqrt(s)*diff into taps
  }
  G[idx] = val;
}

// ---------------------------------------------------------------------------
// Kernel 3: scale-blocked FIR bank as fp32 WMMA matmul, 2x4 tiles per wave.
// Grid: (SIGLEN/512, 4 scale-block pairs, 32 batches); 256 thr = 8 waves.
// Wave w: scales 32q..32q+31 (2 M-tiles), l = l0wg + w*64 .. +63 (4 N-tiles).
// ---------------------------------------------------------------------------
__global__ __launch_bounds__(256)
void cwt_wmma_kernel(const float* __restrict__ x,
                     const float* __restrict__ G,
                     float* __restrict__ out) {
  __shared__ float xs[WGPOS + GSTRIDE + 8];  // shared x window (<= 10.3 KB)

  const int tid  = threadIdx.x;
  const int q    = blockIdx.y;               // scale-block pair: scales 32q+1..32q+32
  const int b    = blockIdx.z;               // batch
  const int l0wg = blockIdx.x * WGPOS;

  const int Kext = 512 * q + 516;            // pair K extent (odd block's), mult of 4
  const int m_lo = -(256 * q + 257);         // first tap offset
  const int nld  = WGPOS + Kext;             // x window length

  // Stage zero-padded x window into LDS.
  const float* xg = x + (size_t)b * SIGLEN;
  for (int i = tid; i < nld; i += 256) {
    int il = l0wg + m_lo + i;
    xs[i] = (il >= 0 && il < SIGLEN) ? xg[il] : 0.0f;
  }
  __syncthreads();

  const int wave = tid >> 5;                 // wave32
  const int lane = tid & 31;
  const int n    = lane & 15;
  const int half = lane >> 4;                // K half for A/B fragments
  const int c0   = 1025 + m_lo;              // first G column (= 768 - 256q >= 0)

  // A fragments: rows 32q+n (M-tile 0) and 32q+16+n (M-tile 1).
  // Lanes 0-15 hold K{0,1}; lanes 16-31 hold K{2,3}.
  const float* ga0 = G + (size_t)(32 * q + n) * GSTRIDE + c0 + half * 2;
  const float* ga1 = ga0 + (size_t)16 * GSTRIDE;
  // B fragments from LDS: N-tile t uses xs[wave*64 + t*16 + n + half*2 + kk + k].
  const int bb = wave * 64 + n + half * 2;

  v8f a00 = {}, a01 = {}, a02 = {}, a03 = {};   // M-tile 0, N-tiles 0..3
  v8f a10 = {}, a11 = {}, a12 = {}, a13 = {};   // M-tile 1, N-tiles 0..3

  for (int kk = 0; kk < Kext; kk += 4) {
    v2f fa0 = *(const v2f*)(ga0 + kk);       // 8B-aligned global loads
    v2f fa1 = *(const v2f*)(ga1 + kk);
    v2f b0, b1, b2, b3;                      // LDS: 4 x ds_load_2addr_b32
    b0.x = xs[bb + kk +  0]; b0.y = xs[bb + kk +  1];
    b1.x = xs[bb + kk + 16]; b1.y = xs[bb + kk + 17];
    b2.x = xs[bb + kk + 32]; b2.y = xs[bb + kk + 33];
    b3.x = xs[bb + kk + 48]; b3.y = xs[bb + kk + 49];

    a00 = __builtin_amdgcn_wmma_f32_16x16x4_f32(false, fa0, false, b0,
                                                (short)0, a00, false, false);
    a01 = __builtin_amdgcn_wmma_f32_16x16x4_f32(false, fa0, false, b1,
                                                (short)0, a01, false, false);
    a02 = __builtin_amdgcn_wmma_f32_16x16x4_f32(false, fa0, false, b2,
                                                (short)0, a02, false, false);
    a03 = __builtin_amdgcn_wmma_f32_16x16x4_f32(false, fa0, false, b3,
                                                (short)0, a03, false, false);
    a10 = __builtin_amdgcn_wmma_f32_16x16x4_f32(false, fa1, false, b0,
                                                (short)0, a10, false, false);
    a11 = __builtin_amdgcn_wmma_f32_16x16x4_f32(false, fa1, false, b1,
                                                (short)0, a11, false, false);
    a12 = __builtin_amdgcn_wmma_f32_16x16x4_f32(false, fa1, false, b2,
                                                (short)0, a12, false, false);
    a13 = __builtin_amdgcn_wmma_f32_16x16x4_f32(false, fa1, false, b3,
                                                (short)0, a13, false, false);
  }

  // D layout: lanes 0-15 hold M=v (VGPR v), lanes 16-31 hold M=v+8; N = lane&15.
  const int srow0 = 32 * q + half * 8;
  float* op0 = out + ((size_t)b * NSCALES + srow0) * (size_t)SIGLEN
                   + l0wg + wave * 64 + n;
  float* op1 = op0 + (size_t)16 * SIGLEN;    // M-tile 1 (scales +16)
  #pragma unroll
  for (int v = 0; v < 8; ++v) {
    op0[(size_t)v * SIGLEN +  0] = a00[v];
    op0[(size_t)v * SIGLEN + 16] = a01[v];
    op0[(size_t)v * SIGLEN + 32] = a02[v];
    op0[(size_t)v * SIGLEN + 48] = a03[v];
    op1[(size_t)v * SIGLEN +  0] = a10[v];
    op1[(size_t)v * SIGLEN + 16] = a11[v];
    op1[(size_t)v * SIGLEN + 32] = a12[v];
    op1[(size_t)v * SIGLEN + 48] = a13[v];
  }
}

// ---------------------------------------------------------------------------
extern "C" void kernel_launch(void* const* d_in, const int* in_sizes, int n_in,
                              void* d_out, int out_size, void* d_ws, size_t ws_size,
                              hipStream_t stream) {
  (void)in_sizes; (void)n_in; (void)out_size; (void)ws_size;

  const float* x  = (const float*)d_in[0];
  float* out      = (float*)d_out;
  float* G        = (float*)d_ws;                     // 128*2052 floats (~1 MB)
  float* int_psi  = G + NSCALES * GSTRIDE;            // + 1024 floats

  cwt_int_psi_kernel<<<1, PREC_N, 0, stream>>>(int_psi);

  const int total = NSCALES * GSTRIDE;
  cwt_build_filters_kernel<<<(total + 255) / 256, 256, 0, stream>>>(int_psi, G);

  dim3 grid(SIGLEN / WGPOS, NSCALES / 32, NBATCH);
  cwt_wmma_kernel<<<grid, 256, 0, stream>>>(x, G, out);
}